// Encoder_31550829756513
// MI455X (gfx1250) — compile-verified
//
#include <hip/hip_runtime.h>
#include <hip/hip_bf16.h>

typedef __attribute__((ext_vector_type(2))) float v2f;
typedef __attribute__((ext_vector_type(8))) float v8f;

#define HID 128
#define OUTF 64

// ---------------- utility kernels ----------------

__global__ void k_zero(float* __restrict__ p, int n) {
    int i = blockIdx.x * blockDim.x + threadIdx.x;
    if (i < n) p[i] = 0.0f;
}

// degree accumulation over rows (both branches) + int64 -> int32 index narrowing
__global__ void k_degree(const long long* __restrict__ ei, const float* __restrict__ ppmi,
                         int E, int* __restrict__ row32, int* __restrict__ col32,
                         float* __restrict__ deg_g, float* __restrict__ deg_p) {
    int e = blockIdx.x * blockDim.x + threadIdx.x;
    if (e >= E) return;
    int r = (int)ei[e];
    int c = (int)ei[(long long)E + e];
    row32[e] = r;
    col32[e] = c;
    atomicAdd(deg_g + r, 1.0f);
    atomicAdd(deg_p + r, ppmi[e]);
}

// dis = (deg + selfloop)^-0.5 ; deg+1 > 0 always so the where() branch is moot
__global__ void k_dis(float* __restrict__ deg_g, float* __restrict__ deg_p, int n) {
    int i = blockIdx.x * blockDim.x + threadIdx.x;
    if (i >= n) return;
    deg_g[i] = rsqrtf(deg_g[i] + 1.0f);
    deg_p[i] = rsqrtf(deg_p[i] + 1.0f);
}

// per-edge symmetric norms for both branches
__global__ void k_norm(const int* __restrict__ row32, const int* __restrict__ col32,
                       const float* __restrict__ ppmi,
                       const float* __restrict__ dis_g, const float* __restrict__ dis_p,
                       float* __restrict__ norm_g, float* __restrict__ norm_p, int E) {
    int e = blockIdx.x * blockDim.x + threadIdx.x;
    if (e >= E) return;
    int r = row32[e], c = col32[e];
    norm_g[e] = dis_g[r] * dis_g[c];
    norm_p[e] = dis_p[r] * ppmi[e] * dis_p[c];
}

// ---------------- WMMA fp32 GEMM: C[M x NF] = A[M x K] * B[K x NF] ----------------
// grid.x = M/16 (M multiple of 16), block = 32*(NF/16) threads, wave w owns cols [16w,16w+16)
template <int K, int NF>
__global__ __launch_bounds__(32 * (NF / 16)) void k_gemm(const float* __restrict__ A,
                                                         const float* __restrict__ B,
                                                         float* __restrict__ C) {
    __shared__ float As[16 * (K + 1)];   // +1 padding: stride 257/129 -> conflict-free banks
    const int tid  = threadIdx.x;
    const int lane = tid & 31;
    const int wave = tid >> 5;
    const int rowBase = blockIdx.x * 16;
    const int nthreads = 32 * (NF / 16);

    // stage 16 x K A-tile into LDS (float4 global loads, scalar LDS stores due to padding)
    for (int c4 = tid; c4 < 16 * K / 4; c4 += nthreads) {
        int r  = (c4 * 4) / K;
        int kk = (c4 * 4) % K;
        const float4 v = *(const float4*)(A + (size_t)(rowBase + r) * K + kk);
        float* dst = As + r * (K + 1) + kk;
        dst[0] = v.x; dst[1] = v.y; dst[2] = v.z; dst[3] = v.w;
    }
    __syncthreads();

    const int m    = lane & 15;            // A row within tile / C col within tile
    const int kq   = (lane >> 4) << 1;     // lanes 0-15 -> K 0,1 ; lanes 16-31 -> K 2,3
    const int ncol = wave * 16 + (lane & 15);

    v8f acc = {0.f, 0.f, 0.f, 0.f, 0.f, 0.f, 0.f, 0.f};
    for (int k0 = 0; k0 < K; k0 += 4) {
        v2f a, b;
        a.x = As[m * (K + 1) + k0 + kq];
        a.y = As[m * (K + 1) + k0 + kq + 1];
        b.x = B[(size_t)(k0 + kq) * NF + ncol];
        b.y = B[(size_t)(k0 + kq + 1) * NF + ncol];
        acc = __builtin_amdgcn_wmma_f32_16x16x4_f32(false, a, false, b,
                                                    (short)0, acc, false, false);
    }

    // C/D layout: VGPR r -> M = r (lanes 0-15) or r+8 (lanes 16-31), N = lane&15
    const int hi = lane >> 4;
#pragma unroll
    for (int r = 0; r < 8; ++r) {
        int mrow = rowBase + r + hi * 8;
        C[(size_t)mrow * NF + ncol] = acc[r];
    }
}

// self-loop initialization of the aggregation accumulators: acc = dis^2 * src
template <int F>
__global__ void k_selfinit(const float* __restrict__ srcg, const float* __restrict__ srcp,
                           const float* __restrict__ dis_g, const float* __restrict__ dis_p,
                           float* __restrict__ dstg, float* __restrict__ dstp, int total) {
    int i = blockIdx.x * blockDim.x + threadIdx.x;
    if (i >= total) return;
    int node = i / F;
    float dg = dis_g[node], dp = dis_p[node];
    dstg[i] = dg * dg * srcg[i];
    dstp[i] = dp * dp * srcp[i];
}

// edge scatter: dst[col] += norm * src[row]; one thread = one edge x 4 features, both branches
template <int F>
__global__ void k_scatter(const int* __restrict__ row32, const int* __restrict__ col32,
                          const float* __restrict__ norm_g, const float* __restrict__ norm_p,
                          const float* __restrict__ srcg, const float* __restrict__ srcp,
                          float* __restrict__ dstg, float* __restrict__ dstp, int E) {
    const int CH = F / 4;
    unsigned gid = blockIdx.x * blockDim.x + threadIdx.x;
    if (gid >= (unsigned)E * CH) return;
    int e  = gid / CH;
    int ch = (gid % CH) * 4;
    int r = row32[e], c = col32[e];
    float ng = norm_g[e], np = norm_p[e];
    const float4 hg = *(const float4*)(srcg + (size_t)r * F + ch);
    const float4 hp = *(const float4*)(srcp + (size_t)r * F + ch);
    float* gd = dstg + (size_t)c * F + ch;
    float* pd = dstp + (size_t)c * F + ch;
    atomicAdd(gd + 0, ng * hg.x); atomicAdd(gd + 1, ng * hg.y);
    atomicAdd(gd + 2, ng * hg.z); atomicAdd(gd + 3, ng * hg.w);
    atomicAdd(pd + 0, np * hp.x); atomicAdd(pd + 1, np * hp.y);
    atomicAdd(pd + 2, np * hp.z); atomicAdd(pd + 3, np * hp.w);
}

// bias + ReLU, in place, both branches (layer-1 hidden width 128)
__global__ void k_biasrelu(float* __restrict__ g, float* __restrict__ p,
                           const float* __restrict__ b1, int total) {
    int i = blockIdx.x * blockDim.x + threadIdx.x;
    if (i >= total) return;
    int f = i & (HID - 1);
    float bb = b1[f];
    g[i] = fmaxf(g[i] + bb, 0.0f);
    p[i] = fmaxf(p[i] + bb, 0.0f);
}

// per-node attention merge: one wave32 per node, 64 features -> 2 per lane
__global__ void k_final(const float* __restrict__ gacc, const float* __restrict__ pacc,
                        const float* __restrict__ b2, const float* __restrict__ dw,
                        const float* __restrict__ db, float* __restrict__ out, int n) {
    int lane = threadIdx.x & 31;
    int node = (blockIdx.x * blockDim.x + threadIdx.x) >> 5;
    if (node >= n) return;
    size_t base = (size_t)node * OUTF;
    float g0 = gacc[base + lane]      + b2[lane];
    float g1 = gacc[base + 32 + lane] + b2[32 + lane];
    float p0 = pacc[base + lane]      + b2[lane];
    float p1 = pacc[base + 32 + lane] + b2[32 + lane];
    float dg = g0 * dw[lane] + g1 * dw[32 + lane];
    float dp = p0 * dw[lane] + p1 * dw[32 + lane];
#pragma unroll
    for (int off = 16; off > 0; off >>= 1) {
        dg += __shfl_xor(dg, off, 32);
        dp += __shfl_xor(dp, off, 32);
    }
    dg += db[0];
    dp += db[0];
    float mx = fmaxf(dg, dp);
    float eg = __expf(dg - mx), ep = __expf(dp - mx);
    float wg = eg / (eg + ep);
    float wp = 1.0f - wg;
    out[base + lane]      = wg * g0 + wp * p0;
    out[base + 32 + lane] = wg * g1 + wp * p1;
}

// ---------------- launch ----------------

extern "C" void kernel_launch(void* const* d_in, const int* in_sizes, int n_in,
                              void* d_out, int out_size, void* d_ws, size_t ws_size,
                              hipStream_t stream) {
    const float*     x     = (const float*)d_in[0];
    const long long* ei    = (const long long*)d_in[1];
    const float*     ppmi  = (const float*)d_in[2];
    const float*     W1    = (const float*)d_in[3];
    const float*     b1    = (const float*)d_in[4];
    const float*     W2    = (const float*)d_in[5];
    const float*     b2    = (const float*)d_in[6];
    const float*     dw    = (const float*)d_in[7];
    const float*     db    = (const float*)d_in[8];
    float*           out   = (float*)d_out;

    const int D = 256;
    const int N = in_sizes[0] / D;      // 100000 (multiple of 16)
    const int E = in_sizes[2];          // 1600000

    // workspace carve-up (all fp32-size aligned; N*4 and E*4 are 16B multiples)
    float* ws    = (float*)d_ws;
    float* deg_g = ws;                 ws += N;          // becomes dis_g
    float* deg_p = ws;                 ws += N;          // becomes dis_p
    float* hw1   = ws;                 ws += (size_t)N * HID;
    float* g1    = ws;                 ws += (size_t)N * HID;
    float* p1    = ws;                 ws += (size_t)N * HID;
    float* g2w   = ws;                 ws += (size_t)N * OUTF;
    float* p2w   = ws;                 ws += (size_t)N * OUTF;
    float* g2a   = ws;                 ws += (size_t)N * OUTF;
    float* p2a   = ws;                 ws += (size_t)N * OUTF;
    float* norm_g = ws;                ws += E;
    float* norm_p = ws;                ws += E;
    int*   row32 = (int*)ws;           ws += E;
    int*   col32 = (int*)ws;           /* end */

    const int T = 256;
    auto cdiv = [](long long a, long long b) { return (unsigned)((a + b - 1) / b); };

    // 1) degrees
    k_zero<<<cdiv(2 * N, T), T, 0, stream>>>(deg_g, 2 * N);   // deg_g,deg_p contiguous
    k_degree<<<cdiv(E, T), T, 0, stream>>>(ei, ppmi, E, row32, col32, deg_g, deg_p);
    k_dis<<<cdiv(N, T), T, 0, stream>>>(deg_g, deg_p, N);
    k_norm<<<cdiv(E, T), T, 0, stream>>>(row32, col32, ppmi, deg_g, deg_p, norm_g, norm_p, E);

    // 2) layer 1: shared GEMM hw1 = x @ W1  (WMMA f32 16x16x4)
    k_gemm<256, HID><<<N / 16, 32 * (HID / 16), 0, stream>>>(x, W1, hw1);

    // 3) layer-1 propagation, both branches
    k_selfinit<HID><<<cdiv((long long)N * HID, T), T, 0, stream>>>(hw1, hw1, deg_g, deg_p,
                                                                   g1, p1, N * HID);
    k_scatter<HID><<<cdiv((long long)E * (HID / 4), T), T, 0, stream>>>(
        row32, col32, norm_g, norm_p, hw1, hw1, g1, p1, E);
    k_biasrelu<<<cdiv((long long)N * HID, T), T, 0, stream>>>(g1, p1, b1, N * HID);

    // 4) layer 2 GEMMs (per branch) + propagation
    k_gemm<HID, OUTF><<<N / 16, 32 * (OUTF / 16), 0, stream>>>(g1, W2, g2w);
    k_gemm<HID, OUTF><<<N / 16, 32 * (OUTF / 16), 0, stream>>>(p1, W2, p2w);
    k_selfinit<OUTF><<<cdiv((long long)N * OUTF, T), T, 0, stream>>>(g2w, p2w, deg_g, deg_p,
                                                                     g2a, p2a, N * OUTF);
    k_scatter<OUTF><<<cdiv((long long)E * (OUTF / 4), T), T, 0, stream>>>(
        row32, col32, norm_g, norm_p, g2w, p2w, g2a, p2a, E);

    // 5) attention merge -> out [N, 64]
    k_final<<<cdiv((long long)N * 32, T), T, 0, stream>>>(g2a, p2a, b2, dw, db, out, N);
}